// ScaleDotProductAttention_69466801045769
// MI455X (gfx1250) — compile-verified
//
#include <hip/hip_runtime.h>

typedef __attribute__((ext_vector_type(2))) float v2f;
typedef __attribute__((ext_vector_type(8))) float v8f;

#define Bx 2
#define Hx 16
#define Lx 2048
#define Dx 64
#define SP 2052   // padded LDS row stride in floats (2048 % 64banks == 0, so pad)

// One workgroup = 4 waves (wave32) = 128 threads; owns a 16-row query tile of one (b,h).
__global__ __launch_bounds__(128)
void sdpa_wmma_f32_kernel(const float* __restrict__ q,
                          const float* __restrict__ k,
                          const float* __restrict__ v,
                          const float* __restrict__ mask,
                          float* __restrict__ out,     // [B,H,L,D]
                          float* __restrict__ score)   // [B,H,L,L]
{
    extern __shared__ float smem[];
    float* sS  = smem;            // 16 x SP  (score/P tile, padded rows)
    float* red = smem + 16 * SP;  // 128 floats reduction scratch

    const int tid  = threadIdx.x;
    const int wave = tid >> 5;
    const int lane = tid & 31;
    const int lo16 = lane & 15;   // M (rows) for A-frag, N (cols) for B-frag / C-frag
    const int hi   = lane >> 4;   // selects K-pair / M-half per WMMA VGPR layout

    const int blk = blockIdx.x;                 // h varies fastest -> mask reuse in L2
    const int h   = blk % Hx;
    const int qt  = (blk / Hx) % (Lx / 16);
    const int bz  = blk / (Hx * (Lx / 16));
    const int q0  = qt * 16;

    const size_t bh = (size_t)(bz * Hx + h);
    const float* qp = q + (bh * Lx + q0) * Dx;
    const float* kp = k + bh * Lx * Dx;
    const float* vp = v + bh * Lx * Dx;
    const float* mp = mask + (size_t)bz * Lx * Lx + (size_t)q0 * Lx;

    // ---- Load Q tile once as 16 WMMA A-fragments (K chunks of 4) ----
    // A 16x4 f32 layout: lanes 0-15 hold M=lane, K={0,1}; lanes 16-31 hold K={2,3}.
    v2f aQ[16];
#pragma unroll
    for (int c = 0; c < 16; ++c) {
        aQ[c] = *(const v2f*)(qp + lo16 * Dx + 4 * c + 2 * hi);
    }

    // ---- Phase 1: S = (Q K^T) * 1/sqrt(D) * mask  -> LDS ----
    for (int t = wave; t < Lx / 16; t += 4) {
        const int n0 = t * 16;
        // B 4x16 f32: lane holds column n=lo16, rows K = 2*hi + {0,1}.
        v2f bK[16];
#pragma unroll
        for (int c = 0; c < 16; ++c) {
            bK[c] = *(const v2f*)(kp + (size_t)(n0 + lo16) * Dx + 4 * c + 2 * hi);
        }
        v8f acc0 = {}, acc1 = {};   // two chains to hide WMMA latency
#pragma unroll
        for (int c = 0; c < 16; c += 2) {
            acc0 = __builtin_amdgcn_wmma_f32_16x16x4_f32(false, aQ[c],     false, bK[c],
                                                         (short)0, acc0, false, false);
            acc1 = __builtin_amdgcn_wmma_f32_16x16x4_f32(false, aQ[c + 1], false, bK[c + 1],
                                                         (short)0, acc1, false, false);
        }
        v8f s = (acc0 + acc1) * 0.125f;   // 1/sqrt(64)
        // C/D layout: VGPR i holds row m = i + 8*hi, col n = lo16.
#pragma unroll
        for (int i = 0; i < 8; ++i) {
            const int m = i + 8 * hi;
            const float mv = mp[(size_t)m * Lx + n0 + lo16];
            sS[m * SP + n0 + lo16] = s[i] * mv;
        }
    }
    __syncthreads();

    // ---- Phase 2: row-wise softmax in LDS; stream normalized P to score (NT) ----
    {
        const int row  = tid >> 3;   // 16 rows
        const int part = tid & 7;    // 8 threads per row
        float* rowp = sS + row * SP;

        float mx = -3.402823466e+38f;
        for (int n = part; n < Lx; n += 8) mx = fmaxf(mx, rowp[n]);
        red[tid] = mx;
        __syncthreads();
#pragma unroll
        for (int j = 0; j < 8; ++j) mx = fmaxf(mx, red[row * 8 + j]);
        __syncthreads();

        float sm = 0.f;
        for (int n = part; n < Lx; n += 8) {
            const float e = __expf(rowp[n] - mx);
            rowp[n] = e;
            sm += e;
        }
        red[tid] = sm;
        __syncthreads();
        float tot = 0.f;
#pragma unroll
        for (int j = 0; j < 8; ++j) tot += red[row * 8 + j];
        const float rn = 1.f / tot;

        float* srow = score + (bh * Lx + q0 + row) * Lx;
        for (int n = part; n < Lx; n += 8) {
            const float p = rowp[n] * rn;
            rowp[n] = p;
            __builtin_nontemporal_store(p, srow + n);   // 512MB write-once stream
        }
    }
    __syncthreads();

    // ---- Phase 3: O = P @ V  (each wave owns 16 of the 64 output columns) ----
    {
        const int n0 = wave * 16;
        v8f oa[4] = {{}, {}, {}, {}};   // 4 chains over the K=2048 reduction
        for (int k0 = 0; k0 < Lx; k0 += 16) {
#pragma unroll
            for (int cc = 0; cc < 4; ++cc) {
                const int kk = k0 + 4 * cc;
                // A-frag from LDS: row m=lo16, K = kk + 2*hi + {0,1}; padded stride => no bank conflict
                v2f a = *(const v2f*)(sS + lo16 * SP + kk + 2 * hi);
                v2f bV;
                bV.x = vp[(size_t)(kk + 2 * hi) * Dx + n0 + lo16];
                bV.y = vp[(size_t)(kk + 2 * hi + 1) * Dx + n0 + lo16];
                oa[cc] = __builtin_amdgcn_wmma_f32_16x16x4_f32(false, a, false, bV,
                                                               (short)0, oa[cc], false, false);
            }
        }
        v8f o = (oa[0] + oa[1]) + (oa[2] + oa[3]);
#pragma unroll
        for (int i = 0; i < 8; ++i) {
            const int m = i + 8 * hi;
            out[(bh * Lx + q0 + m) * Dx + n0 + lo16] = o[i];
        }
    }
}

extern "C" void kernel_launch(void* const* d_in, const int* in_sizes, int n_in,
                              void* d_out, int out_size, void* d_ws, size_t ws_size,
                              hipStream_t stream) {
    const float* q    = (const float*)d_in[0];
    const float* k    = (const float*)d_in[1];
    const float* v    = (const float*)d_in[2];
    const float* mask = (const float*)d_in[3];

    float* out   = (float*)d_out;                       // first output: [B,H,L,D]
    float* score = out + (size_t)Bx * Hx * Lx * Dx;     // second output: [B,H,L,L]

    const int grid = Bx * Hx * (Lx / 16);               // 4096 blocks
    const size_t shmem = (size_t)(16 * SP + 128) * sizeof(float);  // ~129 KB dynamic LDS

    sdpa_wmma_f32_kernel<<<grid, 128, shmem, stream>>>(q, k, v, mask, out, score);
}